// SoftToHardEncoder_27608049779089
// MI455X (gfx1250) — compile-verified
//
#include <hip/hip_runtime.h>

// SoftToHardEncoder: B=16, L=16, H=64, W=64, NUM_CODES=64
// d_in[0] = z       [B, L, H, W] float32  (1,048,576 elems)
// d_in[1] = codes_w [L, 64]      float32  (1,024 elems)
// d_out   = soft[B,H,W,L] ++ hard[B,H,W,L] ++ idx[B,H,W,L]  (3 * 1,048,576 floats)

#define B_DIM 16
#define L_DIM 16
#define H_DIM 64
#define W_DIM 64
#define C_DIM 64
#define CW_STRIDE 68                               // 16B-aligned rows, conflict-free
#define N_SITES (B_DIM * H_DIM * W_DIM * L_DIM)   // 1,048,576

typedef __attribute__((address_space(1))) int gint_t;
typedef __attribute__((address_space(3))) int lint_t;

#if defined(__gfx1250__) && __has_builtin(__builtin_amdgcn_global_load_async_to_lds_b32)
#define USE_ASYNC_LDS 1
#else
#define USE_ASYNC_LDS 0
#endif

#if USE_ASYNC_LDS
__device__ __forceinline__ void async_copy_b32(const float* g, float* l) {
    __builtin_amdgcn_global_load_async_to_lds_b32(
        (gint_t*)(void*)const_cast<float*>(g),   // addrspacecast generic -> global
        (lint_t*)(void*)l,                       // addrspacecast generic -> LDS
        0, 0);
}
#endif

__device__ __forceinline__ void wait_asynccnt0() {
#if __has_builtin(__builtin_amdgcn_s_wait_asynccnt)
    __builtin_amdgcn_s_wait_asynccnt(0);
#else
    asm volatile("s_wait_asynccnt 0" ::: "memory");
#endif
}

__device__ __forceinline__ float fast_exp_neg(float d) {
    // exp(-d), d >= 0; data is N(0,1) so d <= ~15 << 87 -> no underflow concern
#if __has_builtin(__builtin_amdgcn_exp2f)
    return __builtin_amdgcn_exp2f(d * -1.442695040888963f);   // v_exp_f32
#else
    return __expf(-d);
#endif
}

__device__ __forceinline__ float fast_rcp(float x) {
#if __has_builtin(__builtin_amdgcn_rcpf)
    return __builtin_amdgcn_rcpf(x);                          // v_rcp_f32
#else
    return 1.0f / x;
#endif
}

__global__ __launch_bounds__(256) void
SoftToHardEncoder_27608049779089_kernel(const float* __restrict__ z,
                                        const float* __restrict__ codes,
                                        float* __restrict__ soft_out,
                                        float* __restrict__ hard_out,
                                        float* __restrict__ idx_out) {
    // z tile transposed: zt[w_local][l], row stride 17 -> conflict-free b32 reads
    __shared__ float zt[16 * 17];
    // codebook: cw[l][c], row stride 68 -> 16B-aligned rows for ds_load_b128;
    // bank spread (4l + c) % 64 distinct over the 16 l's in a wave.
    __shared__ float cw[L_DIM * CW_STRIDE];

    const int t    = threadIdx.x;          // 0..255
    const int tile = blockIdx.x;           // 4096 blocks
    const int w0   = (tile & 3) << 4;      // 4 w-tiles of 16
    const int h    = (tile >> 2) & (H_DIM - 1);
    const int b    = tile >> 8;

    // ---------------- staging phase (transpose done via LDS dest address) ----
    const int wl_ld = t & 15;              // fast dim -> coalesced global read
    const int l_ld  = t >> 4;
    const float* gz = z + (((size_t)(b * L_DIM + l_ld) * H_DIM + h) * W_DIM + w0 + wl_ld);
    float*       lz = &zt[wl_ld * 17 + l_ld];

#if USE_ASYNC_LDS
    // one B32 async global->LDS copy per lane; LDS address performs the transpose
    async_copy_b32(gz, lz);
#pragma unroll
    for (int i = t; i < L_DIM * C_DIM; i += 256) {
        async_copy_b32(codes + i, &cw[(i >> 6) * CW_STRIDE + (i & 63)]);
    }
    wait_asynccnt0();
#else
    *lz = *gz;
#pragma unroll
    for (int i = t; i < L_DIM * C_DIM; i += 256) {
        cw[(i >> 6) * CW_STRIDE + (i & 63)] = codes[i];
    }
#endif
    __syncthreads();

    // ---------------- compute phase (single fused pass) ----------------------
    const int l  = t & 15;
    const int wl = t >> 4;
    const float zv = zt[wl * 17 + l];
    const float4* __restrict__ crow4 = (const float4*)&cw[l * CW_STRIDE];

    float dmin = __builtin_inff();
    int   imin = 0;
    // 4 independent accumulator pairs to break the FP dependency chain
    float s0 = 0.f, s1 = 0.f, s2 = 0.f, s3 = 0.f;
    float w0a = 0.f, w1a = 0.f, w2a = 0.f, w3a = 0.f;

#pragma unroll
    for (int q = 0; q < C_DIM / 4; ++q) {
        const float4 cv = crow4[q];                // ds_load_b128
        const int c = q << 2;

        float d0 = fabsf(zv - cv.x);
        float d1 = fabsf(zv - cv.y);
        float d2 = fabsf(zv - cv.z);
        float d3 = fabsf(zv - cv.w);

        // first-occurrence argmin (strict <), matching jnp.argmin
        if (d0 < dmin) { dmin = d0; imin = c + 0; }
        if (d1 < dmin) { dmin = d1; imin = c + 1; }
        if (d2 < dmin) { dmin = d2; imin = c + 2; }
        if (d3 < dmin) { dmin = d3; imin = c + 3; }

        float e0 = fast_exp_neg(d0);
        float e1 = fast_exp_neg(d1);
        float e2 = fast_exp_neg(d2);
        float e3 = fast_exp_neg(d3);

        s0 += e0;  w0a = fmaf(e0, cv.x, w0a);
        s1 += e1;  w1a = fmaf(e1, cv.y, w1a);
        s2 += e2;  w2a = fmaf(e2, cv.z, w2a);
        s3 += e3;  w3a = fmaf(e3, cv.w, w3a);
    }

    const float s  = (s0 + s1) + (s2 + s3);
    const float ws = (w0a + w1a) + (w2a + w3a);
    const float* __restrict__ crow = &cw[l * CW_STRIDE];

    // out index: ((b*H + h)*W + (w0 + wl))*L + l == base + t  (coalesced)
    const int out = ((b * H_DIM + h) * W_DIM + w0) * L_DIM + t;
    soft_out[out] = ws * fast_rcp(s);
    hard_out[out] = crow[imin];
    idx_out[out]  = (float)imin;
}

extern "C" void kernel_launch(void* const* d_in, const int* in_sizes, int n_in,
                              void* d_out, int out_size, void* d_ws, size_t ws_size,
                              hipStream_t stream) {
    (void)in_sizes; (void)n_in; (void)out_size; (void)d_ws; (void)ws_size;
    const float* z     = (const float*)d_in[0];
    const float* codes = (const float*)d_in[1];
    float* out  = (float*)d_out;
    float* soft = out;
    float* hard = out + N_SITES;
    float* idx  = out + 2 * N_SITES;

    const int blocks = B_DIM * H_DIM * (W_DIM / 16);   // 4096
    SoftToHardEncoder_27608049779089_kernel<<<blocks, 256, 0, stream>>>(
        z, codes, soft, hard, idx);
}